// EDN_Model_18811956757061
// MI455X (gfx1250) — compile-verified
//
#include <hip/hip_runtime.h>
#include <hip/hip_bf16.h>

#define NN 40000
#define EE 480000
#define LL 40
#define BB 12
#define NCA 10000
#define R1C 5.0f
#define R2C 10.0f

typedef __attribute__((ext_vector_type(2))) float v2f;
typedef __attribute__((ext_vector_type(8))) float v8f;

// ---------------------------------------------------------------------------
// WMMA helper: D = A(16x4,f32) * B(4x16,f32) + C(16x16,f32)
//   A: lane l: row M = l&15 ; a.x = A[M][k0], a.y = A[M][k0+1], k0 = 2*(l>>4)
//   B: lane l: col N = l&15 ; b.x = B[k0][N], b.y = B[k0+1][N]
//   C/D: c[r] = C[M = r + 8*(l>>4)][N = l&15]
// ---------------------------------------------------------------------------
__device__ inline v8f wmma4(v2f a, v2f b, v8f c) {
    return __builtin_amdgcn_wmma_f32_16x16x4_f32(false, a, false, b, (short)0, c,
                                                 false, false);
}

__device__ inline float ssp(float v) {  // shifted softplus
    float sp = (v > 20.f) ? v : log1pf(__expf(v));
    return sp - 0.69314718056f;
}

// ---------------------------------------------------------------------------
// CG table builder (15 threads, doubles, once per launch)
// ---------------------------------------------------------------------------
__constant__ int c_paths[15][3] = {
    {0,0,0},{0,1,1},{0,2,2},{1,0,1},{1,1,0},{1,1,1},{1,1,2},{1,2,1},
    {1,2,2},{2,0,2},{2,1,1},{2,1,2},{2,2,0},{2,2,1},{2,2,2}};
__constant__ int c_cgoff[15] = {0,1,10,35,44,53,80,125,170,245,270,315,390,415,490};

struct cx { double r, i; };
__device__ inline cx cmul(cx a, cx b){ return cx{a.r*b.r - a.i*b.i, a.r*b.i + a.i*b.r}; }
__device__ inline cx cconj(cx a){ return cx{a.r, -a.i}; }
__device__ inline double dfact(int n){ double r = 1.0; for (int i = 2; i <= n; i++) r *= (double)i; return r; }

__device__ void mkU(int l, cx U[5][5]) {
    for (int a = 0; a < 5; a++) for (int b = 0; b < 5; b++) U[a][b] = cx{0, 0};
    const double s2 = 0.7071067811865476;
    U[l][l] = cx{1, 0};
    for (int m = 1; m <= l; m++) {
        double sg = (m & 1) ? -1.0 : 1.0;
        U[l+m][l+m] = cx{sg*s2, 0};
        U[l+m][l-m] = cx{0, sg*s2};
        U[l-m][l+m] = cx{s2, 0};
        U[l-m][l-m] = cx{0, -s2};
    }
}

__global__ void k_build_cg(float* __restrict__ cgbuf) {
    int p = threadIdx.x;
    if (p >= 15) return;
    int j1 = c_paths[p][0], j2 = c_paths[p][1], j3 = c_paths[p][2];
    int n1 = 2*j1+1, n2 = 2*j2+1, n3 = 2*j3+1;
    double C[5][5][5];
    for (int a = 0; a < 5; a++) for (int b = 0; b < 5; b++) for (int c = 0; c < 5; c++) C[a][b][c] = 0.0;
    for (int m1 = -j1; m1 <= j1; m1++) for (int m2 = -j2; m2 <= j2; m2++) {
        int m3 = m1 + m2;
        if (m3 < -j3 || m3 > j3) continue;
        double pre = sqrt((2.0*j3+1.0)*dfact(j1+j2-j3)*dfact(j1-j2+j3)*dfact(-j1+j2+j3)/dfact(j1+j2+j3+1));
        pre *= sqrt(dfact(j3+m3)*dfact(j3-m3)*dfact(j1-m1)*dfact(j1+m1)*dfact(j2-m2)*dfact(j2+m2));
        double s = 0.0;
        for (int k = 0; k <= j1+j2-j3; k++) {
            int den[6] = {k, j1+j2-j3-k, j1-m1-k, j2+m2-k, j3-j2+m1+k, j3-j1-m2+k};
            bool ok = true; double dd = 1.0;
            for (int t = 0; t < 6; t++) { if (den[t] < 0) { ok = false; break; } dd *= dfact(den[t]); }
            if (!ok) continue;
            s += ((k & 1) ? -1.0 : 1.0) / dd;
        }
        C[m1+j1][m2+j2][m3+j3] = pre * s;
    }
    cx U1[5][5], U2[5][5], U3[5][5];
    mkU(j1, U1); mkU(j2, U2); mkU(j3, U3);
    double CrR[5][5][5], CrI[5][5][5];
    double mr = 0, mi = 0;
    for (int A = 0; A < n1; A++) for (int Bq = 0; Bq < n2; Bq++) for (int Cc = 0; Cc < n3; Cc++) {
        cx acc = cx{0, 0};
        for (int a = 0; a < n1; a++) for (int b = 0; b < n2; b++) for (int c = 0; c < n3; c++) {
            double cv = C[a][b][c];
            if (cv == 0.0) continue;
            cx t = cmul(U1[a][A], U2[b][Bq]);
            t = cmul(t, cconj(U3[c][Cc]));
            acc.r += cv * t.r; acc.i += cv * t.i;
        }
        CrR[A][Bq][Cc] = acc.r; CrI[A][Bq][Cc] = acc.i;
        mr = fmax(mr, fabs(acc.r)); mi = fmax(mi, fabs(acc.i));
    }
    bool useI = mi > mr;
    int idx = c_cgoff[p];
    for (int A = 0; A < n1; A++) for (int Bq = 0; Bq < n2; Bq++) for (int Cc = 0; Cc < n3; Cc++)
        cgbuf[idx++] = (float)(useI ? CrI[A][Bq][Cc] : CrR[A][Bq][Cc]);
}

// ---------------------------------------------------------------------------
// Edge geometry: Y1 (3), Y2 (5), Gaussian bases g1/g2 (12 each)
// ---------------------------------------------------------------------------
__global__ void k_geom(const float* __restrict__ ea, float* __restrict__ Y1,
                       float* __restrict__ Y2, float* __restrict__ g1,
                       float* __restrict__ g2) {
    int e = blockIdx.x * blockDim.x + threadIdx.x;
    if (e >= EE) return;
    float x = ea[e*3+0], y = ea[e*3+1], z = ea[e*3+2];
    float r = sqrtf(x*x + y*y + z*z + 1e-12f);
    float inv = 1.f / r;
    x *= inv; y *= inv; z *= inv;
    Y1[e*3+0] = 0.48860251f * y;
    Y1[e*3+1] = 0.48860251f * z;
    Y1[e*3+2] = 0.48860251f * x;
    const float c2 = 1.09254843f;
    Y2[e*5+0] = c2 * x * y;
    Y2[e*5+1] = c2 * y * z;
    Y2[e*5+2] = 0.31539157f * (3.f*z*z - 1.f);
    Y2[e*5+3] = c2 * x * z;
    Y2[e*5+4] = 0.54627422f * (x*x - y*y);
    const float w1 = R1C / (BB - 1), w2 = R2C / (BB - 1);
    #pragma unroll
    for (int k = 0; k < BB; k++) {
        float t1 = (r - (float)k * w1) / w1;
        float t2 = (r - (float)k * w2) / w2;
        g1[e*BB+k] = __expf(-t1*t1);
        g2[e*BB+k] = __expf(-t2*t2);
    }
}

// ---------------------------------------------------------------------------
// Node embedding h = x @ lin1_w^T + b, WMMA. 128 threads = 4 waves; grid exact.
// Weights staged in LDS zero-padded to 48 cols -> unconditional B reads.
// ---------------------------------------------------------------------------
__global__ void k_embed(const float* __restrict__ xin, const float* __restrict__ w,
                        const float* __restrict__ b, float* __restrict__ h) {
    __shared__ float Ws[20][48];
    __shared__ float Bs[48];
    int t = threadIdx.x;
    for (int idx = t; idx < 20*48; idx += 128) {
        int r = idx / 48, c = idx % 48;
        Ws[r][c] = (c < LL) ? w[c*20 + r] : 0.f;     // Ws[k][n] = lin1_w[n][k]
    }
    if (t < 48) Bs[t] = (t < LL) ? b[t] : 0.f;
    __syncthreads();

    int wave = t >> 5;
    int task = blockIdx.x * 4 + wave;                // 1875*4 == 7500, exact
    int nt = task % 3, mt = task / 3;
    int lane = t & 31, lr = lane & 15, half = lane >> 4, k0 = 2 * half;
    int row = mt * 16 + lr;
    int col = nt * 16 + lr;                          // 0..47
    v8f acc;
    float bb = Bs[col];
    #pragma unroll
    for (int r = 0; r < 8; r++) acc[r] = bb;
    #pragma unroll
    for (int kk = 0; kk < 20; kk += 4) {
        v2f a, bv;
        a.x = xin[row*20 + kk + k0];
        a.y = xin[row*20 + kk + k0 + 1];
        bv.x = Ws[kk + k0][col];
        bv.y = Ws[kk + k0 + 1][col];
        acc = wmma4(a, bv, acc);
    }
    #pragma unroll
    for (int r = 0; r < 8; r++) {
        int m = r + 8 * half;
        if (col < LL) h[(mt*16 + m)*LL + col] = acc[r];
    }
}

// ---------------------------------------------------------------------------
// Fused radial-MLP (WMMA) + conv1 scatter.
// 8 waves/block, 16 edges/wave, weights+bias staged zero-padded in LDS.
// After layer2, per-edge weights live in LDS; same wave scatters its 16 edges.
// ---------------------------------------------------------------------------
__global__ void __launch_bounds__(256)
k_rconv1(const float* __restrict__ g, const float* __restrict__ w1,
         const float* __restrict__ b1, const float* __restrict__ w2,
         const float* __restrict__ b2, const float* __restrict__ h,
         const float* __restrict__ Y, const int* __restrict__ src,
         const int* __restrict__ dst, float* __restrict__ out, int m) {
    __shared__ float W1s[12][16];
    __shared__ float B1s[16];
    __shared__ float W2s[12][48];
    __shared__ float B2s[48];
    __shared__ float Hs[8][16][16];
    __shared__ float Ew[8][16][48];   // per-edge radial weights (cols 0..39 valid)

    int t = threadIdx.x;
    for (int idx = t; idx < 12*16; idx += 256) {
        int r = idx >> 4, c = idx & 15;
        W1s[r][c] = (c < BB) ? w1[r*BB + c] : 0.f;
    }
    for (int idx = t; idx < 12*48; idx += 256) {
        int r = idx / 48, c = idx % 48;
        W2s[r][c] = (c < LL) ? w2[r*LL + c] : 0.f;
    }
    if (t < 16) B1s[t] = (t < BB) ? b1[t] : 0.f;
    if (t < 48) B2s[t] = (t < LL) ? b2[t] : 0.f;
    __syncthreads();

    int wave = t >> 5, lane = t & 31, lr = lane & 15, half = lane >> 4, k0 = 2*half;
    int tile = blockIdx.x * 8 + wave;                // grid exact: EE/16/8
    int e0 = tile * 16;

    // layer 1: (16x12) @ (12x12)
    v8f acc;
    float bb = B1s[lr];
    #pragma unroll
    for (int r = 0; r < 8; r++) acc[r] = bb;
    #pragma unroll
    for (int kk = 0; kk < BB; kk += 4) {
        int k = kk + k0;
        v2f a, bv;
        a.x = g[(e0 + lr)*BB + k];
        a.y = g[(e0 + lr)*BB + k + 1];
        bv.x = W1s[k][lr];
        bv.y = W1s[k+1][lr];
        acc = wmma4(a, bv, acc);
    }
    #pragma unroll
    for (int r = 0; r < 8; r++) Hs[wave][r + 8*half][lr] = fmaxf(acc[r], 0.f);

    // layer 2: (16x12) @ (12x40) -> Ew   (same-wave LDS is in-order; barrier for safety)
    __syncthreads();
    #pragma unroll
    for (int ntile = 0; ntile < 3; ntile++) {
        int col = ntile * 16 + lr;                   // 0..47
        v8f a2;
        float b2v = B2s[col];
        #pragma unroll
        for (int r = 0; r < 8; r++) a2[r] = b2v;
        #pragma unroll
        for (int kk = 0; kk < BB; kk += 4) {
            int k = kk + k0;
            v2f a, bv;
            a.x = Hs[wave][lr][k];
            a.y = Hs[wave][lr][k+1];
            bv.x = W2s[k][col];
            bv.y = W2s[k+1][col];
            a2 = wmma4(a, bv, a2);
        }
        #pragma unroll
        for (int r = 0; r < 8; r++) Ew[wave][r + 8*half][col] = a2[r];
    }
    __syncthreads();

    // scatter: CG(0,o,o)=I -> out[dst,c,k] += h[src,c]*Ew[e,c]*Y[e,k]
    for (int ee = 0; ee < 16; ee++) {
        int e = e0 + ee;
        int s = src[e], d = dst[e];
        if (ee + 1 < 16) {
            int sn = src[e + 1];
            __builtin_prefetch(h + (size_t)sn * LL, 0, 0);   // global_prefetch_b8
        }
        float yv[5];
        if (m == 1) yv[0] = 0.28209479f;
        else for (int k = 0; k < m; k++) yv[k] = Y[e*m + k];
        for (int c = lane; c < LL; c += 32) {
            float a = h[s*LL + c] * Ew[wave][ee][c];
            for (int k = 0; k < m; k++)
                atomicAdd(&out[(d*LL + c)*m + k], a * yv[k]);
        }
    }
}

// ---------------------------------------------------------------------------
// Fused radial-MLP (WMMA) + conv2 scatter (ca-only: dst%4==0).
// msg[c,k] = Ew[e,c] * sum_i x[src,c,i]*ycg[i,k],  ycg[i,k]=sum_j cg[i,j,k]*Y_j
// ---------------------------------------------------------------------------
__global__ void __launch_bounds__(256)
k_rconv2(const float* __restrict__ g, const float* __restrict__ w1,
         const float* __restrict__ b1, const float* __restrict__ w2,
         const float* __restrict__ b2, const float* __restrict__ xin,
         const float* __restrict__ Yf, const float* __restrict__ cg,
         const int* __restrict__ src, const int* __restrict__ dst,
         float* __restrict__ out, int mi, int mf, int mo, int coff, int ctot) {
    __shared__ float W1s[12][16];
    __shared__ float B1s[16];
    __shared__ float W2s[12][48];
    __shared__ float B2s[48];
    __shared__ float CGs[128];
    __shared__ float Hs[8][16][16];
    __shared__ float Ew[8][16][48];

    int t = threadIdx.x;
    int csz = mi * mf * mo;
    for (int idx = t; idx < 12*16; idx += 256) {
        int r = idx >> 4, c = idx & 15;
        W1s[r][c] = (c < BB) ? w1[r*BB + c] : 0.f;
    }
    for (int idx = t; idx < 12*48; idx += 256) {
        int r = idx / 48, c = idx % 48;
        W2s[r][c] = (c < LL) ? w2[r*LL + c] : 0.f;
    }
    if (t < 16) B1s[t] = (t < BB) ? b1[t] : 0.f;
    if (t < 48) B2s[t] = (t < LL) ? b2[t] : 0.f;
    if (t < 128) CGs[t] = (t < csz) ? cg[t] : 0.f;
    __syncthreads();

    int wave = t >> 5, lane = t & 31, lr = lane & 15, half = lane >> 4, k0 = 2*half;
    int tile = blockIdx.x * 8 + wave;
    int e0 = tile * 16;

    // layer 1
    v8f acc;
    float bb = B1s[lr];
    #pragma unroll
    for (int r = 0; r < 8; r++) acc[r] = bb;
    #pragma unroll
    for (int kk = 0; kk < BB; kk += 4) {
        int k = kk + k0;
        v2f a, bv;
        a.x = g[(e0 + lr)*BB + k];
        a.y = g[(e0 + lr)*BB + k + 1];
        bv.x = W1s[k][lr];
        bv.y = W1s[k+1][lr];
        acc = wmma4(a, bv, acc);
    }
    #pragma unroll
    for (int r = 0; r < 8; r++) Hs[wave][r + 8*half][lr] = fmaxf(acc[r], 0.f);
    __syncthreads();

    // layer 2 -> Ew
    #pragma unroll
    for (int ntile = 0; ntile < 3; ntile++) {
        int col = ntile * 16 + lr;
        v8f a2;
        float b2v = B2s[col];
        #pragma unroll
        for (int r = 0; r < 8; r++) a2[r] = b2v;
        #pragma unroll
        for (int kk = 0; kk < BB; kk += 4) {
            int k = kk + k0;
            v2f a, bv;
            a.x = Hs[wave][lr][k];
            a.y = Hs[wave][lr][k+1];
            bv.x = W2s[k][col];
            bv.y = W2s[k+1][col];
            a2 = wmma4(a, bv, a2);
        }
        #pragma unroll
        for (int r = 0; r < 8; r++) Ew[wave][r + 8*half][col] = a2[r];
    }
    __syncthreads();

    // scatter (only dst % 4 == 0 survives the [ca] slice)
    for (int ee = 0; ee < 16; ee++) {
        int e = e0 + ee;
        int d = dst[e];
        if (d & 3) continue;
        int s = src[e], d4 = d >> 2;
        __builtin_prefetch(xin + (size_t)s * LL * mi, 0, 0);  // global_prefetch_b8
        float yv[5];
        if (mf == 1) yv[0] = 0.28209479f;
        else for (int j = 0; j < mf; j++) yv[j] = Yf[e*mf + j];
        float ycg[25];
        for (int i = 0; i < mi; i++)
            for (int k = 0; k < mo; k++) {
                float a = 0.f;
                for (int j = 0; j < mf; j++) a += CGs[(i*mf + j)*mo + k] * yv[j];
                ycg[i*mo + k] = a;
            }
        for (int c = lane; c < LL; c += 32) {
            float wc = Ew[wave][ee][c];
            float xv[5];
            for (int i = 0; i < mi; i++) xv[i] = xin[(s*LL + c)*mi + i];
            for (int k = 0; k < mo; k++) {
                float a = 0.f;
                for (int i = 0; i < mi; i++) a += xv[i] * ycg[i*mo + k];
                atomicAdd(&out[((size_t)d4*ctot + coff + c)*mo + k], wc * a);
            }
        }
    }
}

// ---------------------------------------------------------------------------
// norm -> lin2 -> nonlin -> lin3 (per node, per l); 64 threads/block
// ---------------------------------------------------------------------------
__global__ void k_chain(const float* __restrict__ in, const float* __restrict__ w2,
                        const float* __restrict__ b2, const float* __restrict__ w3,
                        const float* __restrict__ b3, float* __restrict__ out, int m) {
    __shared__ float F[LL*5], G[LL*5];
    int n = blockIdx.x, t = threadIdx.x;
    if (t < LL) {
        float f[5], s = 0.f;
        for (int j = 0; j < m; j++) { f[j] = in[(n*LL + t)*m + j]; s += f[j]*f[j]; }
        float inv = 1.f / (sqrtf(s) + 1e-8f);
        for (int j = 0; j < m; j++) F[t*m + j] = f[j] * inv;
    }
    __syncthreads();
    if (t < LL) {
        float gq[5];
        for (int j = 0; j < m; j++) gq[j] = b2 ? b2[t] : 0.f;
        for (int c = 0; c < LL; c++) {
            float wv = w2[t*LL + c];
            for (int j = 0; j < m; j++) gq[j] += wv * F[c*m + j];
        }
        if (m == 1) gq[0] = ssp(gq[0]);
        else {
            float nn = 0.f; for (int j = 0; j < m; j++) nn += gq[j]*gq[j];
            nn = sqrtf(nn + 1e-12f);
            float sc = ssp(nn) / (nn + 1e-8f);
            for (int j = 0; j < m; j++) gq[j] *= sc;
        }
        for (int j = 0; j < m; j++) G[t*m + j] = gq[j];
    }
    __syncthreads();
    if (t < LL) {
        float o[5];
        for (int j = 0; j < m; j++) o[j] = b3 ? b3[t] : 0.f;
        for (int c = 0; c < LL; c++) {
            float wv = w3[t*LL + c];
            for (int j = 0; j < m; j++) o[j] += wv * G[c*m + j];
        }
        for (int j = 0; j < m; j++) out[(n*LL + t)*m + j] = o[j];
    }
}

// ---------------------------------------------------------------------------
// final vector heads (o=1,2): norm_f -> lin -> gated nonlin -> d_out columns
// ---------------------------------------------------------------------------
__global__ void k_fv(const float* __restrict__ tin, const float* __restrict__ w,
                     float* __restrict__ out, int m, int colbase) {
    __shared__ float F[240*5];
    int n = blockIdx.x, t = threadIdx.x;
    for (int c = t; c < 240; c += 64) {
        float f[5], s = 0.f;
        for (int j = 0; j < m; j++) { f[j] = tin[((size_t)n*240 + c)*m + j]; s += f[j]*f[j]; }
        float inv = 1.f / (sqrtf(s) + 1e-8f);
        for (int j = 0; j < m; j++) F[c*m + j] = f[j] * inv;
    }
    __syncthreads();
    if (t < LL) {
        float g[5];
        for (int j = 0; j < m; j++) g[j] = 0.f;
        for (int c = 0; c < 240; c++) {
            float wv = w[t*240 + c];
            for (int j = 0; j < m; j++) g[j] += wv * F[c*m + j];
        }
        float nn = 0.f; for (int j = 0; j < m; j++) nn += g[j]*g[j];
        nn = sqrtf(nn + 1e-12f);
        float sc = ssp(nn) / (nn + 1e-8f);
        for (int j = 0; j < m; j++)
            out[(size_t)n*321 + colbase + t*m + j] = g[j] * sc;
    }
}

// ---------------------------------------------------------------------------
// scalar head: lin40+ssp -> d1+elu -> d2 -> d3 -> d_out col 0
// ---------------------------------------------------------------------------
__global__ void k_f0(const float* __restrict__ t0, const float* __restrict__ w40,
                     const float* __restrict__ b40, const float* __restrict__ d1w,
                     const float* __restrict__ d1b, const float* __restrict__ d2w,
                     const float* __restrict__ d2b, const float* __restrict__ d3w,
                     const float* __restrict__ d3b, float* __restrict__ out) {
    __shared__ float H[LL], A1[250], A2[152];
    int n = blockIdx.x, t = threadIdx.x;
    if (t < LL) {
        float a = b40[t];
        for (int c = 0; c < 120; c++) a += w40[t*120 + c] * t0[(size_t)n*120 + c];
        H[t] = ssp(a);
    }
    __syncthreads();
    for (int o = t; o < 250; o += 256) {
        float a = d1b[o];
        for (int c = 0; c < LL; c++) a += d1w[o*LL + c] * H[c];
        A1[o] = (a > 0.f) ? a : expm1f(a);          // ELU
    }
    __syncthreads();
    for (int o = t; o < 150; o += 256) {
        float a = d2b[o];
        for (int c = 0; c < 250; c++) a += d2w[o*250 + c] * A1[c];
        A2[o] = a;
    }
    __syncthreads();
    if (t == 0) {
        float a = d3b[0];
        for (int c = 0; c < 150; c++) a += d3w[c] * A2[c];
        out[(size_t)n*321] = a;
    }
}

// ---------------------------------------------------------------------------
extern "C" void kernel_launch(void* const* d_in, const int* in_sizes, int n_in,
                              void* d_out, int out_size, void* d_ws, size_t ws_size,
                              hipStream_t stream) {
    const float* x        = (const float*)d_in[0];
    const int*   ei       = (const int*)d_in[1];
    const float* ea       = (const float*)d_in[2];
    const float* lin1_w   = (const float*)d_in[3];
    const float* lin1_b   = (const float*)d_in[4];
    const float* c1rw1    = (const float*)d_in[5];
    const float* c1rb1    = (const float*)d_in[6];
    const float* c1rw2    = (const float*)d_in[7];
    const float* c1rb2    = (const float*)d_in[8];
    const float* lin2_w   = (const float*)d_in[9];
    const float* lin2_b0  = (const float*)d_in[10];
    const float* lin3_w   = (const float*)d_in[11];
    const float* lin3_b0  = (const float*)d_in[12];
    const float* c2rw1    = (const float*)d_in[13];
    const float* c2rb1    = (const float*)d_in[14];
    const float* c2rw2    = (const float*)d_in[15];
    const float* c2rb2    = (const float*)d_in[16];
    const float* lin40_w  = (const float*)d_in[17];
    const float* lin40_b  = (const float*)d_in[18];
    const float* lin41_w  = (const float*)d_in[19];
    const float* lin42_w  = (const float*)d_in[20];
    const float* d1w      = (const float*)d_in[21];
    const float* d1b      = (const float*)d_in[22];
    const float* d2w      = (const float*)d_in[23];
    const float* d2b      = (const float*)d_in[24];
    const float* d3w      = (const float*)d_in[25];
    const float* d3b      = (const float*)d_in[26];
    float* out = (float*)d_out;

    const int* src = ei;
    const int* dst = ei + EE;

    // workspace layout (floats)
    float* p   = (float*)d_ws;
    float* cg  = p;  p += 1024;
    float* Y1  = p;  p += (size_t)EE*3;
    float* Y2  = p;  p += (size_t)EE*5;
    float* g1  = p;  p += (size_t)EE*12;
    float* g2  = p;  p += (size_t)EE*12;
    float* hb  = p;  p += (size_t)NN*LL;
    float* o0  = p;  p += (size_t)NN*LL;
    float* o1  = p;  p += (size_t)NN*LL*3;
    float* o2  = p;  p += (size_t)NN*LL*5;
    float* x0  = p;  p += (size_t)NN*LL;
    float* x1  = p;  p += (size_t)NN*LL*3;
    float* x2  = p;  p += (size_t)NN*LL*5;
    float* t0  = p;  p += (size_t)NCA*120;
    float* t1  = p;  p += (size_t)NCA*720;
    float* t2  = p;  p += (size_t)NCA*1200;

    hipMemsetAsync(o0, 0, (size_t)NN*LL*sizeof(float), stream);
    hipMemsetAsync(o1, 0, (size_t)NN*LL*3*sizeof(float), stream);
    hipMemsetAsync(o2, 0, (size_t)NN*LL*5*sizeof(float), stream);
    hipMemsetAsync(t0, 0, (size_t)NCA*120*sizeof(float), stream);
    hipMemsetAsync(t1, 0, (size_t)NCA*720*sizeof(float), stream);
    hipMemsetAsync(t2, 0, (size_t)NCA*1200*sizeof(float), stream);

    k_build_cg<<<1, 32, 0, stream>>>(cg);
    k_geom<<<EE/256, 256, 0, stream>>>(ea, Y1, Y2, g1, g2);

    // node embedding: (N/16)*3 = 7500 wave tasks, 4 waves/block, grid exact
    k_embed<<<1875, 128, 0, stream>>>(x, lin1_w, lin1_b, hb);

    // conv1: fused radial(WMMA) + scatter, 3 output orders
    float* outs[3] = {o0, o1, o2};
    const float* Ys[3] = {nullptr, Y1, Y2};
    for (int o = 0; o < 3; o++) {
        k_rconv1<<<(EE/16)/8, 256, 0, stream>>>(g1, c1rw1 + o*144, c1rb1 + o*12,
                                                c1rw2 + o*480, c1rb2 + o*40,
                                                hb, Ys[o], src, dst, outs[o], 2*o + 1);
    }

    // per-l channel mixing chain
    float* xl[3] = {x0, x1, x2};
    for (int l = 0; l < 3; l++) {
        k_chain<<<NN, 64, 0, stream>>>(outs[l], lin2_w + l*1600,
                                       l == 0 ? lin2_b0 : nullptr,
                                       lin3_w + l*1600,
                                       l == 0 ? lin3_b0 : nullptr,
                                       xl[l], 2*l + 1);
    }

    // conv2: 15 fused radial+scatter paths (ca-only)
    static const int paths[15][3] = {
        {0,0,0},{0,1,1},{0,2,2},{1,0,1},{1,1,0},{1,1,1},{1,1,2},{1,2,1},
        {1,2,2},{2,0,2},{2,1,1},{2,1,2},{2,2,0},{2,2,1},{2,2,2}};
    static const int cgoff[15] = {0,1,10,35,44,53,80,125,170,245,270,315,390,415,490};
    static const int slot[15]  = {0,0,0,1,1,2,1,3,2,3,4,4,2,5,5};  // index within o-group
    float* tms[3] = {t0, t1, t2};
    const int ctots[3] = {120, 240, 240};
    for (int pth = 0; pth < 15; pth++) {
        int i = paths[pth][0], f = paths[pth][1], o = paths[pth][2];
        k_rconv2<<<(EE/16)/8, 256, 0, stream>>>(g2, c2rw1 + pth*144, c2rb1 + pth*12,
                                                c2rw2 + pth*480, c2rb2 + pth*40,
                                                xl[i], Ys[f], cg + cgoff[pth],
                                                src, dst, tms[o],
                                                2*i + 1, 2*f + 1, 2*o + 1,
                                                slot[pth] * LL, ctots[o]);
    }

    // final heads -> d_out (NCA x 321): [y | out1(120) | out2(200)]
    k_fv<<<NCA, 64, 0, stream>>>(t1, lin41_w, out, 3, 1);
    k_fv<<<NCA, 64, 0, stream>>>(t2, lin42_w, out, 5, 121);
    k_f0<<<NCA, 256, 0, stream>>>(t0, lin40_w, lin40_b, d1w, d1b, d2w, d2b,
                                  d3w, d3b, out);
}